// EnhancedRareVariantFusion_8366596293141
// MI455X (gfx1250) — compile-verified
//
#include <hip/hip_runtime.h>
#include <math.h>

// B=4, K=32, L=512, D=512, H=2048 ; rows = B*L = 2048
#define DD 512
#define HH 2048
#define ROWS 2048

typedef __attribute__((ext_vector_type(2))) float v2f;
typedef __attribute__((ext_vector_type(8))) float v8f;

__device__ __forceinline__ float gelu_f(float x) {
  // exact gelu: x/2 * (1 + erf(x/sqrt(2)))
  return 0.5f * x * (1.0f + erff(x * 0.70710678118654752440f));
}
__device__ __forceinline__ float sigmoid_f(float x) {
  return 1.0f / (1.0f + expf(-x));
}

// ---- block (256 threads = 8 wave32) sum reduction ----
__device__ __forceinline__ float block_reduce_sum_256(float v, float* lds8) {
#pragma unroll
  for (int o = 16; o > 0; o >>= 1) v += __shfl_xor(v, o, 32);
  const int wid = threadIdx.x >> 5;
  const int lane = threadIdx.x & 31;
  if (lane == 0) lds8[wid] = v;
  __syncthreads();
  float r = (lane < 8) ? lds8[lane] : 0.0f;
  if (wid == 0) {
#pragma unroll
    for (int o = 4; o > 0; o >>= 1) r += __shfl_xor(r, o, 32);
    if (lane == 0) lds8[0] = r;
  }
  __syncthreads();
  float out = lds8[0];
  __syncthreads();
  return out;
}

// =====================================================================
// WMMA fp32 GEMM: C[M,N] = act(A[M,K] @ B[K,N] + bias[N])
// Each wave computes a 32x32 macro-tile = 2x2 WMMA 16x16 tiles, so
// 2 A-fragments + 2 B-fragments feed 4 v_wmma ops (2x fragment reuse).
// Block = 8 waves arranged 2(M) x 4(N) -> 64 x 128 block tile.
// grid = (N/128, M/64). EPI: 0=none, 1=gelu, 2=sigmoid
// V_WMMA_F32_16X16X4_F32 fragment layout (ISA 7.12.2):
//   A 16x4 : lane%16 = M row; VGPR0 = K=2*(lane/16), VGPR1 = K=2*(lane/16)+1
//   B 4x16 : lane%16 = N col; VGPR0 = K=2*(lane/16), VGPR1 = K=2*(lane/16)+1
//   C 16x16: VGPR r -> row r + 8*(lane/16), col = lane%16
// =====================================================================
template <int KDIM, int NDIM, int EPI>
__global__ __launch_bounds__(256) void gemm_wmma_f32(
    const float* __restrict__ A, const float* __restrict__ Bw,
    const float* __restrict__ bias, float* __restrict__ C) {
  const int lane = threadIdx.x & 31;
  const int wave = threadIdx.x >> 5;
  const int half = lane >> 4;
  const int r16 = lane & 15;
  const int mBase = blockIdx.y * 64 + (wave >> 2) * 32;
  const int nBase = blockIdx.x * 128 + (wave & 3) * 32;

  const float* __restrict__ Ar0 = A + (size_t)(mBase + r16) * KDIM;
  const float* __restrict__ Ar1 = A + (size_t)(mBase + 16 + r16) * KDIM;
  const float* __restrict__ Bc = Bw + nBase + r16;

  v8f acc00 = {}, acc01 = {}, acc10 = {}, acc11 = {};
#pragma unroll 2
  for (int k0 = 0; k0 < KDIM; k0 += 4) {
    const int ka = k0 + 2 * half;
    v2f a0, a1, b0, b1;
    a0.x = Ar0[ka];
    a0.y = Ar0[ka + 1];
    a1.x = Ar1[ka];
    a1.y = Ar1[ka + 1];
    const float* __restrict__ Bp = Bc + (size_t)ka * NDIM;
    b0.x = Bp[0];
    b0.y = Bp[NDIM];
    b1.x = Bp[16];
    b1.y = Bp[NDIM + 16];
    acc00 = __builtin_amdgcn_wmma_f32_16x16x4_f32(false, a0, false, b0,
                                                  (short)0, acc00, false, false);
    acc01 = __builtin_amdgcn_wmma_f32_16x16x4_f32(false, a0, false, b1,
                                                  (short)0, acc01, false, false);
    acc10 = __builtin_amdgcn_wmma_f32_16x16x4_f32(false, a1, false, b0,
                                                  (short)0, acc10, false, false);
    acc11 = __builtin_amdgcn_wmma_f32_16x16x4_f32(false, a1, false, b1,
                                                  (short)0, acc11, false, false);
  }
#pragma unroll
  for (int r = 0; r < 8; ++r) {
    const int row0 = mBase + half * 8 + r;
    const int row1 = row0 + 16;
    const int col0 = nBase + r16;
    const int col1 = col0 + 16;
    float v00 = acc00[r] + bias[col0];
    float v01 = acc01[r] + bias[col1];
    float v10 = acc10[r] + bias[col0];
    float v11 = acc11[r] + bias[col1];
    if (EPI == 1) {
      v00 = gelu_f(v00); v01 = gelu_f(v01);
      v10 = gelu_f(v10); v11 = gelu_f(v11);
    }
    if (EPI == 2) {
      v00 = sigmoid_f(v00); v01 = sigmoid_f(v01);
      v10 = sigmoid_f(v10); v11 = sigmoid_f(v11);
    }
    C[(size_t)row0 * NDIM + col0] = v00;
    C[(size_t)row0 * NDIM + col1] = v01;
    C[(size_t)row1 * NDIM + col0] = v10;
    C[(size_t)row1 * NDIM + col1] = v11;
  }
}

// ---- column sums: csum[j] = sum_i<512 lda_w[i,j] ; csum2[n] = sum_{i=512..1023} fus_w1[i,n]
__global__ void k_colsums(const float* __restrict__ lda_w,
                          const float* __restrict__ fus_w1,
                          float* __restrict__ csum, float* __restrict__ csum2) {
  const int tid = blockIdx.x * 256 + threadIdx.x;
  if (tid < DD) {
    float s = 0.f;
    for (int i = 0; i < DD; ++i) s += lda_w[(size_t)i * DD + tid];
    csum[tid] = s;
  } else if (tid < DD + HH) {
    const int n = tid - DD;
    float s = 0.f;
    for (int i = DD; i < 2 * DD; ++i) s += fus_w1[(size_t)i * HH + n];
    csum2[n] = s;
  }
}

// ---- CrossAFInteraction -> fused_af[row, d] ; one block per row (b,l) ----
__global__ __launch_bounds__(256) void k_fusedaf(
    const float* __restrict__ gaf, const float* __restrict__ paf,
    const float* __restrict__ gw1, const float* __restrict__ gb1,
    const float* __restrict__ gw2, const float* __restrict__ gb2,
    const float* __restrict__ jw, const float* __restrict__ jb,
    const float* __restrict__ jg, const float* __restrict__ jbe,
    const float* __restrict__ aw, const float* __restrict__ ab,
    const float* __restrict__ res1p, float* __restrict__ out) {
  __shared__ float h_s[32];
  __shared__ float red[8];
  const int row = blockIdx.x;
  const float ga = fminf(fmaxf(gaf[row], 1e-4f), 0.9999f);
  const float pa = fminf(fmaxf(paf[row], 1e-4f), 0.9999f);
  if (threadIdx.x < 32) {
    const int j = threadIdx.x;
    h_s[j] = gelu_f(ga * gw1[j] + pa * gw1[32 + j] + gb1[j]);
  }
  __syncthreads();
  float encl[2], gate[2];
  float sum = 0.f, sumsq = 0.f;
#pragma unroll
  for (int t = 0; t < 2; ++t) {
    const int d = threadIdx.x + t * 256;
    float acc = gb2[d];
#pragma unroll
    for (int j = 0; j < 32; ++j) acc += h_s[j] * gw2[j * DD + d];
    gate[t] = sigmoid_f(acc);
    const float e = ga * jw[d] + pa * jw[DD + d] + jb[d];
    encl[t] = e;
    sum += e;
    sumsq += e * e;
  }
  sum = block_reduce_sum_256(sum, red);
  sumsq = block_reduce_sum_256(sumsq, red);
  const float mean = sum * (1.f / DD);
  const float rstd = rsqrtf(sumsq * (1.f / DD) - mean * mean + 1e-5f);
  const float r1 = res1p[0];
#pragma unroll
  for (int t = 0; t < 2; ++t) {
    const int d = threadIdx.x + t * 256;
    const float enc = gelu_f((encl[t] - mean) * rstd * jg[d] + jbe[d]);
    out[(size_t)row * DD + d] = ga * aw[d] + ab[d] + r1 * (gate[t] * enc);
  }
}

// ---- per-row LayerNorm then gelu, in place (pos_emb) ----
__global__ __launch_bounds__(256) void k_ln_then_gelu(
    float* __restrict__ X, const float* __restrict__ g,
    const float* __restrict__ be) {
  __shared__ float red[8];
  const size_t base = (size_t)blockIdx.x * DD;
  const float v0 = X[base + threadIdx.x];
  const float v1 = X[base + threadIdx.x + 256];
  float sum = block_reduce_sum_256(v0 + v1, red);
  float sumsq = block_reduce_sum_256(v0 * v0 + v1 * v1, red);
  const float mean = sum * (1.f / DD);
  const float rstd = rsqrtf(sumsq * (1.f / DD) - mean * mean + 1e-5f);
  X[base + threadIdx.x] =
      gelu_f((v0 - mean) * rstd * g[threadIdx.x] + be[threadIdx.x]);
  X[base + threadIdx.x + 256] =
      gelu_f((v1 - mean) * rstd * g[threadIdx.x + 256] + be[threadIdx.x + 256]);
}

// ---- S[b,k,l] = sum_d rag[b,k,l,d] * af_w[b,l,d] * pos_e[b,l,d] ----
// one wave per (b,k,l) row; block = 8 rows
__global__ __launch_bounds__(256) void k_S(const float* __restrict__ rag,
                                           const float* __restrict__ afw,
                                           const float* __restrict__ pose,
                                           float* __restrict__ S) {
  const int wave = threadIdx.x >> 5, lane = threadIdx.x & 31;
  const size_t row = (size_t)blockIdx.x * 8 + wave;  // over B*K*L = 65536
  const int b = (int)(row >> 14);                    // / (K*L)
  const int l = (int)(row & 511);
  const size_t bl = (size_t)b * 512 + l;
  const float* __restrict__ rp = rag + row * DD;
  const float* __restrict__ ap = afw + bl * DD;
  const float* __restrict__ pp = pose + bl * DD;
  float s = 0.f;
#pragma unroll
  for (int i = 0; i < 16; ++i) {
    const int j = lane + i * 32;
    s += rp[j] * ap[j] * pp[j];
  }
#pragma unroll
  for (int o = 16; o > 0; o >>= 1) s += __shfl_xor(s, o, 32);
  if (lane == 0) S[row] = s;
}

// ---- scores over K, softmax over K, P[b,l] = sum_k aw_k * S[b,k,l] ----
// one block per (b,l); wave handles one k at a time (8 k's in parallel)
__global__ __launch_bounds__(256) void k_scores(
    const float* __restrict__ S, const float* __restrict__ csum,
    const float* __restrict__ q /* O2 + lda_b */,
    const float* __restrict__ orig, const float* __restrict__ ldag,
    const float* __restrict__ ldabe, float* __restrict__ P) {
  __shared__ float o_s[DD], q_s[DD], c_s[DD], g_s[DD], be_s[DD];
  __shared__ float sc_s[32];
  const int row = blockIdx.x;  // b*L + l
  const int b = row >> 9, l = row & 511;
  for (int j = threadIdx.x; j < DD; j += 256) {
    o_s[j] = orig[(size_t)row * DD + j];
    q_s[j] = q[(size_t)row * DD + j];
    c_s[j] = csum[j];
    g_s[j] = ldag[j];
    be_s[j] = ldabe[j];
  }
  __syncthreads();
  const int wave = threadIdx.x >> 5, lane = threadIdx.x & 31;
  for (int kk = 0; kk < 4; ++kk) {
    const int kidx = kk * 8 + wave;
    const float Sv = S[((size_t)b * 32 + kidx) * 512 + l];
    float garr[16];
    float sum = 0.f, sumsq = 0.f;
#pragma unroll
    for (int i = 0; i < 16; ++i) {
      const int j = lane + i * 32;
      const float g = gelu_f(Sv * c_s[j] + q_s[j]);
      garr[i] = g;
      sum += g;
      sumsq += g * g;
    }
#pragma unroll
    for (int o = 16; o > 0; o >>= 1) {
      sum += __shfl_xor(sum, o, 32);
      sumsq += __shfl_xor(sumsq, o, 32);
    }
    const float mean = sum * (1.f / DD);
    const float rstd = rsqrtf(sumsq * (1.f / DD) - mean * mean + 1e-5f);
    float dot = 0.f;
#pragma unroll
    for (int i = 0; i < 16; ++i) {
      const int j = lane + i * 32;
      dot += ((garr[i] - mean) * rstd * g_s[j] + be_s[j]) * o_s[j];
    }
#pragma unroll
    for (int o = 16; o > 0; o >>= 1) dot += __shfl_xor(dot, o, 32);
    if (lane == 0) sc_s[kidx] = dot * 0.04419417382415921659f;  // 1/sqrt(512)
  }
  __syncthreads();
  if (wave == 0) {
    const float sc = sc_s[lane];
    float m = sc;
#pragma unroll
    for (int o = 16; o > 0; o >>= 1) m = fmaxf(m, __shfl_xor(m, o, 32));
    const float e = expf(sc - m);
    float es = e;
#pragma unroll
    for (int o = 16; o > 0; o >>= 1) es += __shfl_xor(es, o, 32);
    const float Sv = S[((size_t)b * 32 + lane) * 512 + l];
    float pv = (e / es) * Sv;
#pragma unroll
    for (int o = 16; o > 0; o >>= 1) pv += __shfl_xor(pv, o, 32);
    if (lane == 0) P[row] = pv;
  }
}

// ---- Hi = gelu(U + P[row]*csum2[n]) in place ----
__global__ void k_hi(float* __restrict__ U, const float* __restrict__ P,
                     const float* __restrict__ cs2) {
  const size_t idx = (size_t)blockIdx.x * 256 + threadIdx.x;
  const int row = (int)(idx >> 11);  // / 2048
  const int n = (int)(idx & 2047);
  U[idx] = gelu_f(U[idx] + P[row] * cs2[n]);
}

// ---- final: out = orig + res_scale * (LN(F)*g+be) * maf_w ----
__global__ __launch_bounds__(256) void k_final(
    const float* __restrict__ F, const float* __restrict__ g,
    const float* __restrict__ be, const float* __restrict__ orig,
    const float* __restrict__ gaf, const float* __restrict__ mafs,
    const float* __restrict__ ress, float* __restrict__ out) {
  __shared__ float red[8];
  const int row = blockIdx.x;
  const size_t base = (size_t)row * DD;
  const float v0 = F[base + threadIdx.x];
  const float v1 = F[base + threadIdx.x + 256];
  const float sum = block_reduce_sum_256(v0 + v1, red);
  const float sumsq = block_reduce_sum_256(v0 * v0 + v1 * v1, red);
  const float mean = sum * (1.f / DD);
  const float rstd = rsqrtf(sumsq * (1.f / DD) - mean * mean + 1e-5f);
  const float ga = gaf[row];
  const float maf = fmaxf(fminf(ga, 1.f - ga), 1e-4f);
  const float mw = fminf(mafs[0] / maf, 5.f);
  const float rs = ress[0];
  {
    const int d = threadIdx.x;
    const float f = (v0 - mean) * rstd * g[d] + be[d];
    out[base + d] = orig[base + d] + rs * f * mw;
  }
  {
    const int d = threadIdx.x + 256;
    const float f = (v1 - mean) * rstd * g[d] + be[d];
    out[base + d] = orig[base + d] + rs * f * mw;
  }
}

extern "C" void kernel_launch(void* const* d_in, const int* in_sizes, int n_in,
                              void* d_out, int out_size, void* d_ws,
                              size_t ws_size, hipStream_t stream) {
  const float* orig = (const float*)d_in[0];
  const float* rag = (const float*)d_in[1];
  const float* gaf = (const float*)d_in[2];
  const float* paf = (const float*)d_in[3];
  // d_in[4] = snp_positions: mathematically dead (softmax rows sum to 1)
  const float* gate_w1 = (const float*)d_in[5];
  const float* gate_b1 = (const float*)d_in[6];
  const float* gate_w2 = (const float*)d_in[7];
  const float* gate_b2 = (const float*)d_in[8];
  const float* joint_w = (const float*)d_in[9];
  const float* joint_b = (const float*)d_in[10];
  const float* joint_g = (const float*)d_in[11];
  const float* joint_be = (const float*)d_in[12];
  const float* afad1_w = (const float*)d_in[13];
  const float* afad1_b = (const float*)d_in[14];
  const float* res1 = (const float*)d_in[15];
  const float* afw_w1 = (const float*)d_in[16];
  const float* afw_b1 = (const float*)d_in[17];
  const float* afw_w2 = (const float*)d_in[18];
  const float* afw_b2 = (const float*)d_in[19];
  const float* pos_w = (const float*)d_in[20];
  const float* pos_b = (const float*)d_in[21];
  const float* pos_g = (const float*)d_in[22];
  const float* pos_be = (const float*)d_in[23];
  const float* lda_w = (const float*)d_in[24];
  const float* lda_b = (const float*)d_in[25];
  const float* lda_g = (const float*)d_in[26];
  const float* lda_be = (const float*)d_in[27];
  const float* fus_w1 = (const float*)d_in[28];
  const float* fus_b1 = (const float*)d_in[29];
  const float* fus_w2 = (const float*)d_in[30];
  const float* fus_b2 = (const float*)d_in[31];
  const float* fus_g = (const float*)d_in[32];
  const float* fus_be = (const float*)d_in[33];
  const float* maf_scale = (const float*)d_in[34];
  const float* res_scale = (const float*)d_in[35];
  float* out = (float*)d_out;

  // ---- workspace layout (floats) ----
  float* ws = (float*)d_ws;
  float* fused_af = ws;                    // 2048*512
  float* af_w = fused_af + ROWS * DD;      // 2048*512
  float* pos_e = af_w + ROWS * DD;         // 2048*512
  float* O2 = pos_e + ROWS * DD;           // 2048*512 (q = orig@lda_w[D:] + lda_b)
  float* F = O2 + ROWS * DD;               // 2048*512
  float* S = F + ROWS * DD;                // 65536
  float* P = S + 65536;                    // 2048
  float* csum = P + ROWS;                  // 512
  float* csum2 = csum + DD;                // 2048
  float* HB = csum2 + HH;                  // 2048*2048 (shared: Hbuf then U)

  // 1) column sums (dead-aggregated-half folds)
  k_colsums<<<10, 256, 0, stream>>>(lda_w, fus_w1, csum, csum2);
  // 2) CrossAFInteraction
  k_fusedaf<<<ROWS, 256, 0, stream>>>(gaf, paf, gate_w1, gate_b1, gate_w2,
                                      gate_b2, joint_w, joint_b, joint_g,
                                      joint_be, afad1_w, afad1_b, res1,
                                      fused_af);
  // 3) af_weight MLP: [2048,512]x[512,2048] gelu -> [2048,2048]x[2048,512] sigmoid
  gemm_wmma_f32<DD, HH, 1>
      <<<dim3(HH / 128, ROWS / 64), 256, 0, stream>>>(fused_af, afw_w1, afw_b1, HB);
  gemm_wmma_f32<HH, DD, 2>
      <<<dim3(DD / 128, ROWS / 64), 256, 0, stream>>>(HB, afw_w2, afw_b2, af_w);
  // 4) pos_emb = gelu(LN(orig @ pos_w + b))
  gemm_wmma_f32<DD, DD, 0>
      <<<dim3(DD / 128, ROWS / 64), 256, 0, stream>>>(orig, pos_w, pos_b, pos_e);
  k_ln_then_gelu<<<ROWS, 256, 0, stream>>>(pos_e, pos_g, pos_be);
  // 5) q = orig @ lda_w[512:,:] + lda_b   (the k-independent half of ctx @ lda_w)
  gemm_wmma_f32<DD, DD, 0><<<dim3(DD / 128, ROWS / 64), 256, 0, stream>>>(
      orig, lda_w + (size_t)DD * DD, lda_b, O2);
  // 6) S[b,k,l] = <rag row, af_w * pos_e>
  k_S<<<65536 / 8, 256, 0, stream>>>(rag, af_w, pos_e, S);
  // 7) scores over K -> softmax -> P[b,l]
  k_scores<<<ROWS, 256, 0, stream>>>(S, csum, O2, orig, lda_g, lda_be, P);
  // 8) fusion: U = orig @ fus_w1[:512,:] + fus_b1 ; Hi = gelu(U + P*csum2)
  gemm_wmma_f32<DD, HH, 0>
      <<<dim3(HH / 128, ROWS / 64), 256, 0, stream>>>(orig, fus_w1, fus_b1, HB);
  k_hi<<<(ROWS * HH) / 256, 256, 0, stream>>>(HB, P, csum2);
  gemm_wmma_f32<HH, DD, 0>
      <<<dim3(DD / 128, ROWS / 64), 256, 0, stream>>>(HB, fus_w2, fus_b2, F);
  // 9) final LN + MAF-scaled residual
  k_final<<<ROWS, 256, 0, stream>>>(F, fus_g, fus_be, orig, gaf, maf_scale,
                                    res_scale, out);
}